// Vectorize_36060545417381
// MI455X (gfx1250) — compile-verified
//
#include <hip/hip_runtime.h>
#include <hip/hip_bf16.h>
#include <math.h>

// Problem constants (DIM=3, N=30)
#define KKC   181        // K = 2*DIM*N+1
#define PP    24360      // P = 30*29*28
#define DEMB  190        // DIM*DIM + K
#define KPAD  192        // padded GEMM inner dim (mult of 4)
#define MP    24368      // P padded to mult of 16 (1523*16)
#define MTILES 1523
#define NTILES 12        // 192/16
#define SORTN 32768      // 2^15 >= P, bitonic size
#define FILLN (SORTN - PP)   // 8408

typedef __attribute__((ext_vector_type(2))) float v2f;
typedef __attribute__((ext_vector_type(8))) float v8f;

// ---------------------------------------------------------------------------
// Kernel 0: G = matrix^T * matrix (30x30), and Bt = zero-padded Ws_out
// (192x192, row-major by n): Bt[n][d] = Ws_out[n][d] for n<181, d<190 else 0.
// N-major layout => per-lane WMMA B fragment (2 consecutive K at fixed N) is
// one contiguous global_load_b64, same pattern as the A fragment.
// ---------------------------------------------------------------------------
__global__ __launch_bounds__(256) void prep_kernel(
    const float* __restrict__ matrix, const float* __restrict__ Ws_out,
    float* __restrict__ G, float* __restrict__ Bt) {
  int t = blockIdx.x * blockDim.x + threadIdx.x;
  if (t < 900) {
    int i = t / 30, j = t % 30;
    float acc = 0.f;
    #pragma unroll
    for (int d = 0; d < 3; ++d) acc += matrix[d * 30 + i] * matrix[d * 30 + j];
    G[t] = acc;
  }
  if (t < KPAD * KPAD) {
    int n = t / KPAD, d = t % KPAD;
    Bt[t] = (n < KKC && d < DEMB) ? Ws_out[n * DEMB + d] : 0.f;
  }
}

// ---------------------------------------------------------------------------
// Kernel 1: per-permutation vec row (gram_flat ++ inner embed), P blocks.
// Decodes lexicographic permutation index -> (a,b,c); rest = sorted complement.
// proj[i][m] = G[comb_i][rest_m]; per-thread k: 27 values, register bitonic
// sort (padded to 32 with +INF), dot with Wd_in row.
// ---------------------------------------------------------------------------
__global__ __launch_bounds__(192) void vec_kernel(
    const float* __restrict__ G, const float* __restrict__ Ws_in,
    const float* __restrict__ Wd_in, float* __restrict__ vec) {
  __shared__ float proj[3][27];
  __shared__ float gram[9];

  int p = blockIdx.x;
  int t = threadIdx.x;

  // decode permutation index (itertools.permutations lexicographic order)
  int i0 = p / 812;            // 812 = 29*28
  int r  = p - i0 * 812;
  int i1 = r / 28;
  int i2 = r - i1 * 28;
  int a = i0;
  int b = i1 + (i1 >= a ? 1 : 0);
  int mn = a < b ? a : b;
  int mx = a < b ? b : a;
  int c = i2;
  if (c >= mn) c++;
  if (c >= mx) c++;
  // sorted removed triple
  int s0 = mn < c ? mn : c;
  int s2 = mx > c ? mx : c;
  int s1 = a + b + c - s0 - s2;

  if (t < 81) {
    int i = t / 27, m = t % 27;
    int ci = (i == 0) ? a : ((i == 1) ? b : c);
    int v = m;                       // m-th element of sorted complement
    if (v >= s0) v++;
    if (v >= s1) v++;
    if (v >= s2) v++;
    proj[i][m] = G[ci * 30 + v];
  } else if (t < 90) {
    int idx = t - 81;
    int i = idx / 3, j = idx % 3;
    int ci = (i == 0) ? a : ((i == 1) ? b : c);
    int cj = (j == 0) ? a : ((j == 1) ? b : c);
    gram[idx] = G[ci * 30 + cj];     // symmetric => matches transpose flatten
  }
  __syncthreads();

  float* vrow = vec + (size_t)p * KPAD;
  if (t < 9) vrow[t] = gram[t];
  if (t == 181 || t == 182) vrow[9 + t] = 0.f;   // zero the K-pad columns

  if (t < KKC) {
    float w0 = Ws_in[t * 3 + 0], w1 = Ws_in[t * 3 + 1], w2 = Ws_in[t * 3 + 2];
    float s[32];
    #pragma unroll
    for (int m = 0; m < 27; ++m)
      s[m] = proj[0][m] * w0 + proj[1][m] * w1 + proj[2][m] * w2;
    #pragma unroll
    for (int m = 27; m < 32; ++m) s[m] = __builtin_inff();

    // fully unrolled 32-wide bitonic sort (ascending) — stays in registers
    #pragma unroll
    for (int kk = 2; kk <= 32; kk <<= 1) {
      #pragma unroll
      for (int j = kk >> 1; j > 0; j >>= 1) {
        #pragma unroll
        for (int i = 0; i < 32; ++i) {
          int l = i ^ j;
          if (l > i) {
            bool up = ((i & kk) == 0);
            float x = s[i], y = s[l];
            float lo = fminf(x, y), hi = fmaxf(x, y);
            s[i] = up ? lo : hi;
            s[l] = up ? hi : lo;
          }
        }
      }
    }
    float acc = 0.f;
    #pragma unroll
    for (int m = 0; m < 27; ++m) acc += Wd_in[t * 27 + m] * s[m];
    vrow[9 + t] = acc;
  }
}

// ---------------------------------------------------------------------------
// Kernel 2: sm2 = vec(24368x192) * Ws_out^T via V_WMMA_F32_16X16X4_F32.
// One wave per 16x16 tile; 48 K-steps of 4. Both A and B fragments are
// contiguous b64 loads per lane. Output scattered column-major into the
// sort buffer: sortbuf[k*SORTN + p].
// ---------------------------------------------------------------------------
__global__ __launch_bounds__(32) void gemm_kernel(
    const float* __restrict__ vec, const float* __restrict__ Bt,
    float* __restrict__ sortbuf) {
  int mt = blockIdx.x, nt = blockIdx.y;
  int lane = threadIdx.x;
  int half = lane >> 4;    // K-pair select (K+0/1 vs K+2/3)
  int l = lane & 15;       // A: row M; B: col N

  const float* Arow = vec + (size_t)(mt * 16 + l) * KPAD + 2 * half;
  const float* Brow = Bt  + (size_t)(nt * 16 + l) * KPAD + 2 * half;

  v8f acc = {};
  #pragma unroll 4
  for (int kk = 0; kk < 48; ++kk) {
    v2f av, bv;
    av.x = Arow[kk * 4];
    av.y = Arow[kk * 4 + 1];
    bv.x = Brow[kk * 4];
    bv.y = Brow[kk * 4 + 1];
    acc = __builtin_amdgcn_wmma_f32_16x16x4_f32(
        /*neg_a=*/false, av, /*neg_b=*/false, bv,
        /*c_mod=*/(short)0, acc, /*reuse_a=*/false, /*reuse_b=*/false);
  }

  int kcol = nt * 16 + l;
  if (kcol < KKC) {
    // D layout: lanes 0-15 hold M=vgpr, lanes 16-31 hold M=vgpr+8, N=l
    float* col = sortbuf + (size_t)kcol * SORTN + mt * 16 + half * 8;
    #pragma unroll
    for (int v = 0; v < 8; ++v) col[v] = acc[v];
  }
}

// ---------------------------------------------------------------------------
// Kernel 3: pad each column [P, SORTN) with +INF (sorts to the top).
// ---------------------------------------------------------------------------
__global__ __launch_bounds__(256) void fill_kernel(float* __restrict__ sortbuf) {
  int t = blockIdx.x * blockDim.x + threadIdx.x;
  if (t >= KKC * FILLN) return;
  int col = t / FILLN;
  int off = t - col * FILLN;
  sortbuf[(size_t)col * SORTN + PP + off] = __builtin_inff();
}

// ---------------------------------------------------------------------------
// Kernel 4: one bitonic compare-exchange stage across all 181 columns.
// ---------------------------------------------------------------------------
__global__ __launch_bounds__(256) void bitonic_stage(
    float* __restrict__ sortbuf, int j, int k) {
  int t = blockIdx.x * blockDim.x + threadIdx.x;
  if (t >= KKC * SORTN) return;
  int col = t >> 15;
  int i = t & (SORTN - 1);
  int l = i ^ j;
  if (l <= i) return;
  float* p = sortbuf + ((size_t)col << 15);
  float x = p[i], y = p[l];
  bool up = ((i & k) == 0);
  bool sw = up ? (x > y) : (x < y);
  if (sw) { p[i] = y; p[l] = x; }
}

// ---------------------------------------------------------------------------
// Kernel 5: out[k] = sum_p Wd_out[k,p] * sorted[p,k]
// ---------------------------------------------------------------------------
__global__ __launch_bounds__(256) void reduce_kernel(
    const float* __restrict__ Wd_out, const float* __restrict__ sortbuf,
    float* __restrict__ out) {
  int k = blockIdx.x;
  int t = threadIdx.x;
  const float* col = sortbuf + (size_t)k * SORTN;
  const float* w = Wd_out + (size_t)k * PP;
  float acc = 0.f;
  for (int p = t; p < PP; p += 256) acc += w[p] * col[p];
  __shared__ float red[256];
  red[t] = acc;
  __syncthreads();
  for (int st = 128; st > 0; st >>= 1) {
    if (t < st) red[t] += red[t + st];
    __syncthreads();
  }
  if (t == 0) out[k] = red[0];
}

// ---------------------------------------------------------------------------
extern "C" void kernel_launch(void* const* d_in, const int* in_sizes, int n_in,
                              void* d_out, int out_size, void* d_ws, size_t ws_size,
                              hipStream_t stream) {
  const float* matrix = (const float*)d_in[0];
  const float* Ws_in  = (const float*)d_in[1];
  const float* Wd_in  = (const float*)d_in[2];
  const float* Ws_out = (const float*)d_in[3];
  const float* Wd_out = (const float*)d_in[4];
  float* out = (float*)d_out;

  float* ws  = (float*)d_ws;
  float* G   = ws;                               // 900 (round 1024)
  float* Bt  = ws + 1024;                        // 192*192
  float* vec = Bt + KPAD * KPAD;                 // 24368*192
  float* sb  = vec + (size_t)MP * KPAD;          // 181*32768
  // total ~= 42.6 MB of d_ws

  prep_kernel<<<144, 256, 0, stream>>>(matrix, Ws_out, G, Bt);
  vec_kernel<<<PP, 192, 0, stream>>>(G, Ws_in, Wd_in, vec);
  gemm_kernel<<<dim3(MTILES, NTILES), 32, 0, stream>>>(vec, Bt, sb);

  int fillN = KKC * FILLN;
  fill_kernel<<<(fillN + 255) / 256, 256, 0, stream>>>(sb);

  int sortBlocks = (KKC * SORTN + 255) / 256;
  for (int k = 2; k <= SORTN; k <<= 1)
    for (int j = k >> 1; j > 0; j >>= 1)
      bitonic_stage<<<sortBlocks, 256, 0, stream>>>(sb, j, k);

  reduce_kernel<<<KKC, 256, 0, stream>>>(Wd_out, sb, out);
}